// Sparse_attention_90907277787364
// MI455X (gfx1250) — compile-verified
//
#include <hip/hip_runtime.h>
#include <hip/hip_bf16.h>
#include <math.h>

// -------- problem constants (from reference) --------
#define COLS     4096
#define TOPK1    6          // TOP_K + 1
#define EPS_F    1e-7f
#define THREADS  256        // 8 wave32 per row
#define NWAVES   (THREADS / 32)
#define VPT      16         // values per thread = COLS / THREADS

typedef unsigned int u32;
typedef u32   __attribute__((ext_vector_type(4))) u32x4;
typedef int   __attribute__((ext_vector_type(4))) i32x4;
typedef int   __attribute__((ext_vector_type(8))) i32x8;
typedef float __attribute__((ext_vector_type(4))) f32x4;

// ---------------- wave32 reductions ----------------
__device__ __forceinline__ float wave_max_f(float v) {
#pragma unroll
    for (int off = 16; off > 0; off >>= 1)
        v = fmaxf(v, __shfl_xor(v, off, 32));
    return v;
}
__device__ __forceinline__ float wave_sum_f(float v) {
#pragma unroll
    for (int off = 16; off > 0; off >>= 1)
        v += __shfl_xor(v, off, 32);
    return v;
}

// Block-wide reductions (256 threads = 8 waves), broadcast result to all lanes.
__device__ __forceinline__ float block_max_f(float v, float* red) {
    v = wave_max_f(v);
    const int wid = threadIdx.x >> 5;
    if ((threadIdx.x & 31) == 0) red[wid] = v;
    __syncthreads();
    float r = red[0];
#pragma unroll
    for (int i = 1; i < NWAVES; ++i) r = fmaxf(r, red[i]);
    __syncthreads();
    return r;
}
__device__ __forceinline__ float block_sum_f(float v, float* red) {
    v = wave_sum_f(v);
    const int wid = threadIdx.x >> 5;
    if ((threadIdx.x & 31) == 0) red[wid] = v;
    __syncthreads();
    float r = red[0];
#pragma unroll
    for (int i = 1; i < NWAVES; ++i) r += red[i];
    __syncthreads();
    return r;
}

// ---------------- kernel: one row per block ----------------
__global__ __launch_bounds__(THREADS) void sparse_attn_topk_norm(
        const float* __restrict__ in, float* __restrict__ out) {
    __shared__ float smem[COLS];          // 16 KB row tile, filled by TDM
    __shared__ float red[NWAVES];         // reduction scratch

    const int tid = threadIdx.x;
    const unsigned long long row = blockIdx.x;
    const unsigned long long gaddr =
        (unsigned long long)(const void*)(in + row * (unsigned long long)COLS);

    // ---- TDM: DMA one 4096-float row (global -> LDS), issued by wave 0 ----
    if (tid < 32) {
        // LDS aperture: wave-relative LDS byte offset == low 32 bits of flat addr
        const u32 lds_off = (u32)(unsigned long long)(const void*)smem;

        // D# group 0: count=1 | lds_addr | global_addr[56:0] | type=2
        u32x4 g0;
        g0.x = 1u;                                   // count = 1 valid descriptor
        g0.y = lds_off;                              // lds_addr (bytes)
        g0.z = (u32)(gaddr & 0xFFFFFFFFull);         // global_addr[31:0]
        g0.w = (u32)((gaddr >> 32) & 0x01FFFFFFull)  // global_addr[56:32]
             | (2u << 30);                           // type = 2 ("image")

        // D# group 1
        i32x8 g1;
        g1[0] = (int)(2u << 16);       // workgroup_mask=0, data_size=2 (4 bytes)
        g1[1] = (int)(4096u << 16);    // tensor_dim0[15:0] in [31:16] (=4096)
        g1[2] = (int)(1u << 16);       // tensor_dim0 hi=0 | tensor_dim1 lo16 = 1
        g1[3] = (int)(4096u << 16);    // tensor_dim1 hi=0 | tile_dim0 = 4096
        g1[4] = 1;                     // tile_dim1 = 1, tile_dim2 = 0
        g1[5] = 4096;                  // tensor_dim0_stride lo32 (elements)
        g1[6] = (int)(4096u << 16);    // dim0_stride hi=0 | tensor_dim1_stride lo16
        g1[7] = 0;                     // tensor_dim1_stride hi

        // D# group 2: tensor_dim2=1, tensor_dim3=1, stride2=0, tile_dim3=0
        i32x4 g2; g2[0] = 1; g2[1] = 1; g2[2] = 0; g2[3] = 0;
        // D# group 3: stride3=0, tensor_dim4=1 (bits[79:48]), tile_dim4=0
        i32x4 g3; g3[0] = 0; g3[1] = (int)(1u << 16); g3[2] = 0; g3[3] = 0;
        // Extra (6-arg toolchain) group: zero-filled, conservative.
        i32x8 g4; g4[0] = 0; g4[1] = 0; g4[2] = 0; g4[3] = 0;
                  g4[4] = 0; g4[5] = 0; g4[6] = 0; g4[7] = 0;

        __builtin_amdgcn_tensor_load_to_lds(g0, g1, g2, g3, g4, /*cpol=*/0);
        __builtin_amdgcn_s_wait_tensorcnt(0);
    }
    __syncthreads();   // publish TDM-written LDS to all 8 waves

    // ---- cache 16 elements/thread in registers (4x ds_load_b128) ----
    f32x4 v4[4];
    const f32x4* sm4 = (const f32x4*)smem;
#pragma unroll
    for (int k = 0; k < 4; ++k) v4[k] = sm4[k * THREADS + tid];
    float* vf = (float*)v4;   // 16 registers after full unroll

    // ---- exact 6th-largest (duplicate-aware order statistic) ----
    // Walk distinct values in descending order, accumulating multiplicities;
    // terminates in <= 6 iterations since each distinct value adds >= 1.
    float thresh = __builtin_inff();
    float cum = 0.0f;
#pragma unroll 1
    for (int it = 0; it < TOPK1; ++it) {
        float lm = -__builtin_inff();
#pragma unroll
        for (int j = 0; j < VPT; ++j) {
            const float x = vf[j];
            if (x < thresh) lm = fmaxf(lm, x);
        }
        const float m = block_max_f(lm, red);   // largest value < thresh (uniform)

        float lc = 0.0f;
#pragma unroll
        for (int j = 0; j < VPT; ++j) lc += (vf[j] == m) ? 1.0f : 0.0f;
        cum += block_sum_f(lc, red);            // uniform across block

        thresh = m;
        if (cum >= (float)TOPK1) break;         // uniform break
    }

    // ---- threshold, normalize, store (non-temporal: write-once stream) ----
    const float delta = thresh - EPS_F;
    float lsum = 0.0f;
#pragma unroll
    for (int j = 0; j < VPT; ++j) {
        const float w = fmaxf(vf[j] - delta, 0.0f);
        vf[j] = w;
        lsum += w;
    }
    const float s = block_sum_f(lsum, red) + EPS_F;
#pragma unroll
    for (int j = 0; j < VPT; ++j) vf[j] = vf[j] / s;   // match ref elementwise div

    // NT stores keep the 128 MB input resident in the 192 MB L2 across graph
    // replays instead of letting the 128 MB output stream evict it.
    f32x4* out4 = (f32x4*)(out + row * (unsigned long long)COLS);
#pragma unroll
    for (int k = 0; k < 4; ++k)
        __builtin_nontemporal_store(v4[k], &out4[k * THREADS + tid]);
}

// ---------------- host-side launch ----------------
extern "C" void kernel_launch(void* const* d_in, const int* in_sizes, int n_in,
                              void* d_out, int out_size, void* d_ws, size_t ws_size,
                              hipStream_t stream) {
    (void)n_in; (void)out_size; (void)d_ws; (void)ws_size;
    const float* attn_s = (const float*)d_in[0];
    float* out = (float*)d_out;
    const int rows = in_sizes[0] / COLS;   // 8192
    sparse_attn_topk_norm<<<rows, THREADS, 0, stream>>>(attn_s, out);
}